// WindowBoundaryHeadWithHints_13477607375029
// MI455X (gfx1250) — compile-verified
//
#include <hip/hip_runtime.h>
#include <hip/hip_bf16.h>
#include <math.h>

// Problem constants (from reference)
#define BB    8
#define SS    4096
#define HH    2048
#define HINTN 4
#define WSZ   16
#define STRD  8
#define NW    512            // windows per batch (range(0,4096,8) all kept)
#define KDIM  2052           // H + HINT
#define MLPH  64
#define MROWS (BB*NW)        // 4096 pooled rows

typedef float v2f __attribute__((ext_vector_type(2)));
typedef float v8f __attribute__((ext_vector_type(8)));

// ---------------------------------------------------------------------------
// Kernel 1: masked-mean pooling over 16-position windows.
// One block per (b,w). 512 threads * float4 = 2048 hidden dims, coalesced.
// Writes pooled[b*NW+w][0..2051] (hint dims at 2048..2051) and window_mask.
// ---------------------------------------------------------------------------
__global__ __launch_bounds__(512) void pool_kernel(
    const float* __restrict__ hidden,   // [B,S,H]
    const int*   __restrict__ amask,    // [B,S]
    const float* __restrict__ hint,     // [B,S,HINT]
    float*       __restrict__ pooled,   // [B*NW, KDIM]
    float*       __restrict__ out_mask) // [B*NW]
{
    __shared__ float mk[WSZ];
    __shared__ int   ic[WSZ];
    __shared__ float sdenom;

    const int blk   = blockIdx.x;
    const int b     = blk / NW;
    const int w     = blk % NW;
    const int start = w * STRD;
    const int t     = threadIdx.x;

    if (t < WSZ) {
        int idx   = start + t;
        int valid = (idx < SS);
        int idxc  = valid ? idx : (SS - 1);
        ic[t] = idxc;
        mk[t] = valid ? (float)amask[b * SS + idxc] : 0.0f;
    }
    __syncthreads();
    if (t == 0) {
        float s = 0.0f;
        #pragma unroll
        for (int k = 0; k < WSZ; ++k) s += mk[k];
        out_mask[b * NW + w] = (s > 0.0f) ? 1.0f : 0.0f;
        sdenom = fmaxf(s, 1.0f);
    }
    __syncthreads();

    const float inv = 1.0f / sdenom;
    const int h0 = t * 4;                       // 512 threads * 4 = 2048 dims
    const float* hb = hidden + (size_t)b * SS * HH;

    float4 acc = make_float4(0.f, 0.f, 0.f, 0.f);
    #pragma unroll
    for (int k = 0; k < WSZ; ++k) {
        const float m = mk[k];
        if (m != 0.0f) {                        // wave-uniform branch
            const float4 v = *(const float4*)(hb + (size_t)ic[k] * HH + h0);
            acc.x += m * v.x; acc.y += m * v.y;
            acc.z += m * v.z; acc.w += m * v.w;
        }
    }

    float* prow = pooled + (size_t)(b * NW + w) * KDIM;
    float4 r = make_float4(acc.x * inv, acc.y * inv, acc.z * inv, acc.w * inv);
    *(float4*)(prow + h0) = r;                  // row stride 2052*4B = 16B-aligned

    if (t < HINTN) {                            // 4 hint dims
        const float* fb = hint + (size_t)b * SS * HINTN;
        float a = 0.0f;
        #pragma unroll
        for (int k = 0; k < WSZ; ++k) a += mk[k] * fb[(size_t)ic[k] * HINTN + t];
        prow[HH + t] = a * inv;
    }
}

// ---------------------------------------------------------------------------
// Kernel 2: h = gelu(pooled @ W1 + b1); logits = h @ W2 + b2
// One wave (32 lanes) per 16-row chunk; 4 fp32 WMMA accumulators cover N=64.
// V_WMMA_F32_16X16X4_F32 fragment layouts (ISA 7.12.2, 32-bit A 16x4):
//   A: VGPR0 lanes0-15 K=0 / lanes16-31 K=2 ; VGPR1 K=1 / K=3 ; M = lane%16
//   B: VGPR0 lanes0-15 K=0 / lanes16-31 K=2 ; VGPR1 K=1 / K=3 ; N = lane%16
//   C/D: VGPR j : lanes0-15 M=j, lanes16-31 M=j+8 ; N = lane%16
// ---------------------------------------------------------------------------
__device__ __forceinline__ float gelu_exact(float x) {
    return 0.5f * x * (1.0f + erff(x * 0.70710678118654752440f));
}

__global__ __launch_bounds__(32) void mlp_kernel(
    const float* __restrict__ pooled,  // [MROWS, KDIM]
    const float* __restrict__ W1,      // [KDIM, MLPH]
    const float* __restrict__ b1,      // [MLPH]
    const float* __restrict__ W2,      // [MLPH]
    const float* __restrict__ b2,      // [1]
    float*       __restrict__ out_logits) // [MROWS]
{
    __shared__ float hsm[16][MLPH + 1];   // +1 pad: no LDS bank conflicts

    const int lane = threadIdx.x;
    const int r0   = blockIdx.x * 16;
    const int m    = lane & 15;
    const int half = lane >> 4;           // 0 -> K{+0,+1}, 1 -> K{+2,+3}

    const float* Arow = pooled + (size_t)(r0 + m) * KDIM + 2 * half;

    v8f acc[4];
    #pragma unroll
    for (int nt = 0; nt < 4; ++nt) acc[nt] = (v8f){0,0,0,0,0,0,0,0};

    for (int kk = 0; kk < KDIM; kk += 4) {
        v2f a;
        a.x = Arow[kk];
        a.y = Arow[kk + 1];

        const float* Bb = W1 + (size_t)(kk + 2 * half) * MLPH + m;
        #pragma unroll
        for (int nt = 0; nt < 4; ++nt) {
            v2f bf;
            bf.x = Bb[nt * 16];            // row K = kk + 2*half
            bf.y = Bb[MLPH + nt * 16];     // row K = kk + 2*half + 1
            acc[nt] = __builtin_amdgcn_wmma_f32_16x16x4_f32(
                /*neg_a=*/false, a, /*neg_b=*/false, bf,
                /*c_mod=*/(short)0, acc[nt],
                /*reuse_a=*/false, /*reuse_b=*/false);
        }
    }

    // bias + exact GELU, scatter the 16x64 tile into LDS
    #pragma unroll
    for (int nt = 0; nt < 4; ++nt) {
        const float bias = b1[nt * 16 + m];
        #pragma unroll
        for (int j = 0; j < 8; ++j) {
            float x = acc[nt][j] + bias;
            hsm[j + 8 * half][nt * 16 + m] = gelu_exact(x);
        }
    }
    __syncthreads();

    // h @ W2 + b2 : 16 lanes, one output row each
    if (lane < 16) {
        float s = 0.0f;
        #pragma unroll
        for (int n = 0; n < MLPH; ++n) s += hsm[lane][n] * W2[n];
        out_logits[r0 + lane] = s + b2[0];
    }
}

// ---------------------------------------------------------------------------
extern "C" void kernel_launch(void* const* d_in, const int* in_sizes, int n_in,
                              void* d_out, int out_size, void* d_ws, size_t ws_size,
                              hipStream_t stream) {
    const float* hidden = (const float*)d_in[0];
    const int*   amask  = (const int*)  d_in[1];
    const float* hint   = (const float*)d_in[2];
    const float* W1     = (const float*)d_in[3];
    const float* b1     = (const float*)d_in[4];
    const float* W2     = (const float*)d_in[5];
    const float* b2     = (const float*)d_in[6];
    float* out    = (float*)d_out;              // [0..4095] logits, [4096..8191] mask
    float* pooled = (float*)d_ws;               // 4096 x 2052 fp32 = 33.6 MB

    pool_kernel<<<BB * NW, 512, 0, stream>>>(hidden, amask, hint, pooled, out + MROWS);
    mlp_kernel<<<MROWS / 16, 32, 0, stream>>>(pooled, W1, b1, W2, b2, out);
}